// SO3_Rotation_84748294684829
// MI455X (gfx1250) — compile-verified
//
#include <hip/hip_runtime.h>

typedef __attribute__((ext_vector_type(2))) float v2f;
typedef __attribute__((ext_vector_type(8))) float v8f;

// Problem constants (lmax=6, mmax=2, C=64 per reference)
constexpr int kS  = 49;   // (lmax+1)^2
constexpr int kM  = 29;   // masked coefficient count for mmax=2
constexpr int kC  = 64;   // channels
constexpr int kKP = 52;   // padded K for GEMM1 (49 -> 52)

// LDS leading dims (picked to avoid 64-bank conflicts)
constexpr int WROT_LD = 52;  // w_rot  [32 x 52]
constexpr int EMB_LD  = 72;  // emb    [52 x 72]
constexpr int WINV_LD = 36;  // w_inv  [64 x 36]
constexpr int ROT_LD  = 72;  // rotated[32 x 72]

// mask: global coeff index for each masked (l,m), |m|<=2
__device__ __constant__ int c_mask[32] = {
    0,
    1, 2, 3,
    4, 5, 6, 7, 8,
    10, 11, 12, 13, 14,
    18, 19, 20, 21, 22,
    28, 29, 30, 31, 32,
    40, 41, 42, 43, 44,
    0, 0, 0};

// l of each masked index
__device__ __constant__ int c_lofm[32] = {
    0, 1, 1, 1, 2, 2, 2, 2, 2, 3, 3, 3, 3, 3,
    4, 4, 4, 4, 4, 5, 5, 5, 5, 5, 6, 6, 6, 6, 6, 0, 0, 0};

// l of each full coeff index s (0..48)
__device__ __constant__ int c_lofs[52] = {
    0,
    1, 1, 1,
    2, 2, 2, 2, 2,
    3, 3, 3, 3, 3, 3, 3,
    4, 4, 4, 4, 4, 4, 4, 4, 4,
    5, 5, 5, 5, 5, 5, 5, 5, 5, 5, 5,
    6, 6, 6, 6, 6, 6, 6, 6, 6, 6, 6, 6, 6,
    0, 0, 0};

// sqrt((2l+1)/(2*mmax+1)) for l>mmax else 1
__device__ __constant__ float c_resc[7] = {
    1.0f, 1.0f, 1.0f,
    1.1832159566f,   // sqrt(7/5)
    1.3416407865f,   // sqrt(9/5)
    1.4832396974f,   // sqrt(11/5)
    1.6124515497f};  // sqrt(13/5)

__global__ __launch_bounds__(256)
void so3_rotate_inv_kernel(const float* __restrict__ wigner,
                           const float* __restrict__ embedding,
                           float* __restrict__ out)
{
    __shared__ float lds_wrot[32 * WROT_LD];   // w_rot padded 32x52
    __shared__ float lds_emb [kKP * EMB_LD];   // embedding padded 52x64 (ld 72)
    __shared__ float lds_winv[64 * WINV_LD];   // w_inv padded 64x32 (ld 36)
    __shared__ float lds_rot [32 * ROT_LD];    // rotated padded 32x64 (ld 72)

    const int tid = threadIdx.x;
    const int e   = blockIdx.x;

    const float* wg = wigner    + (size_t)e * (kS * kS);
    const float* em = embedding + (size_t)e * (kS * kC);
    float*       op = out       + (size_t)e * (kS * kC);

    // ---- Phase 1: stage w_rot (masked wigner rows) and embedding into LDS
    for (int idx = tid; idx < 32 * WROT_LD; idx += 256) {
        int m = idx / WROT_LD;
        int s = idx - m * WROT_LD;
        float v = 0.0f;
        if (m < kM && s < kS) v = wg[c_mask[m] * kS + s];
        lds_wrot[idx] = v;
    }
    for (int idx = tid; idx < kKP * kC; idx += 256) {
        int s = idx >> 6;          // / 64
        int c = idx & 63;
        lds_emb[s * EMB_LD + c] = (s < kS) ? em[idx] : 0.0f;
    }
    __syncthreads();

    // ---- Phase 2a: derive rescaled w_inv[s][m] from LDS (single HBM read of wigner)
    for (int idx = tid; idx < 64 * 32; idx += 256) {
        int s = idx >> 5;
        int m = idx & 31;
        float v = 0.0f;
        if (s < kS && m < kM) {
            int ls = c_lofs[s];
            float f = (ls == c_lofm[m]) ? c_resc[ls] : 1.0f;
            v = lds_wrot[m * WROT_LD + s] * f;
        }
        lds_winv[s * WINV_LD + m] = v;
    }

    const int wave  = tid >> 5;
    const int lane  = tid & 31;
    const int lhalf = lane >> 4;       // 0: lanes 0-15, 1: lanes 16-31
    const int l15   = lane & 15;
    const int khalf = lhalf * 2;       // A/B fragment K sub-offset

    // ---- Phase 2b: GEMM1  rotated[32x64] = w_rot[32x52] x emb[52x64]
    // 8 tiles of 16x16, one per wave.
    {
        const int tm   = wave >> 2;            // 0..1
        const int tn   = wave & 3;             // 0..3
        const int mrow = tm * 16 + l15;
        const int ncol = tn * 16 + l15;
        v8f acc = {};
        for (int k = 0; k < kKP; k += 4) {
            v2f a = *(const v2f*)&lds_wrot[mrow * WROT_LD + k + khalf];
            v2f b;
            b.x = lds_emb[(k + khalf    ) * EMB_LD + ncol];
            b.y = lds_emb[(k + khalf + 1) * EMB_LD + ncol];
            acc = __builtin_amdgcn_wmma_f32_16x16x4_f32(
                /*neg_a=*/false, a, /*neg_b=*/false, b,
                /*c_mod=*/(short)0, acc, /*reuse_a=*/false, /*reuse_b=*/false);
        }
        const int crow = tm * 16 + lhalf * 8;
        const int ccol = tn * 16 + l15;
#pragma unroll
        for (int r = 0; r < 8; ++r)
            lds_rot[(crow + r) * ROT_LD + ccol] = acc[r];
    }
    __syncthreads();

    // ---- Phase 3: GEMM2  out[64x64] = w_inv[64x32] x rotated[32x64]
    // 16 tiles of 16x16, two per wave; rows >= 49 discarded at store.
    for (int t = wave; t < 16; t += 8) {
        const int sm   = t >> 2;               // 0..3
        const int sn   = t & 3;                // 0..3
        const int mrow = sm * 16 + l15;
        const int ncol = sn * 16 + l15;
        v8f acc = {};
        for (int k = 0; k < 32; k += 4) {
            v2f a = *(const v2f*)&lds_winv[mrow * WINV_LD + k + khalf];
            v2f b;
            b.x = lds_rot[(k + khalf    ) * ROT_LD + ncol];
            b.y = lds_rot[(k + khalf + 1) * ROT_LD + ncol];
            acc = __builtin_amdgcn_wmma_f32_16x16x4_f32(
                false, a, false, b, (short)0, acc, false, false);
        }
        const int srow = sm * 16 + lhalf * 8;
        const int scol = sn * 16 + l15;
#pragma unroll
        for (int r = 0; r < 8; ++r) {
            int s = srow + r;
            if (s < kS) op[s * kC + scol] = acc[r];
        }
    }
}

extern "C" void kernel_launch(void* const* d_in, const int* in_sizes, int n_in,
                              void* d_out, int out_size, void* d_ws, size_t ws_size,
                              hipStream_t stream) {
    const float* wigner    = (const float*)d_in[0];
    const float* embedding = (const float*)d_in[1];
    float*       out       = (float*)d_out;

    const int E = in_sizes[0] / (kS * kS);   // 40000 for the reference setup
    so3_rotate_inv_kernel<<<E, 256, 0, stream>>>(wigner, embedding, out);
}